// MessagePassing_45414984188550
// MI455X (gfx1250) — compile-verified
//
#include <hip/hip_runtime.h>

#define D_FEAT 64
#define EPB 128            // edges per block
#define THREADS 256        // 8 wave32 per block
#define LANES_PER_EDGE 16  // each lane handles 4 floats (one float4) of the 64-wide row

// ---------------------------------------------------------------------------
// Kernel 1: zero the output (harness poisons d_out with 0xAA).
// ---------------------------------------------------------------------------
__global__ __launch_bounds__(THREADS) void gnn_zero_kernel(float4* __restrict__ out, int n4) {
    int i = blockIdx.x * blockDim.x + threadIdx.x;
    if (i < n4) out[i] = make_float4(0.f, 0.f, 0.f, 0.f);
}

// ---------------------------------------------------------------------------
// Kernel 2: gather + scatter-add.
//   - Stage this block's int64 edge indices into LDS with CDNA5 async DMA
//     (global_load_async_to_lds_b64, tracked by ASYNCcnt).
//   - 16 lanes cooperate on one edge: coalesced 256B row gather (b128/lane),
//     4x global_atomic_add_f32 per lane into the target row.
// ---------------------------------------------------------------------------
__global__ __launch_bounds__(THREADS) void gnn_scatter_add_kernel(
    const float* __restrict__ x,
    const long long* __restrict__ eidx,   // [2, E] int64: row 0 = src, row 1 = dst
    float* __restrict__ out,
    int E) {
    __shared__ long long s_idx[2 * EPB];  // [0..EPB) = src, [EPB..2*EPB) = dst

    const int tid = threadIdx.x;
    const long long e0 = (long long)blockIdx.x * EPB;

    // ---- async stage of indices into LDS (one b64 per lane) ----
    {
        // tid in [0,EPB): source indices; tid in [EPB,2*EPB): target indices
        const int local = (tid < EPB) ? tid : (tid - EPB);
        const long long gelem = (tid < EPB) ? (e0 + local) : ((long long)E + e0 + local);
        const long long* gptr = eidx + gelem;
        // low 32 bits of a generic pointer to LDS == LDS byte offset
        unsigned lds_addr = (unsigned)(unsigned long long)(&s_idx[tid]);
        if (e0 + local < E) {
            asm volatile("global_load_async_to_lds_b64 %0, %1, off"
                         :
                         : "v"(lds_addr), "v"(gptr)
                         : "memory");
        }
        asm volatile("s_wait_asynccnt 0" ::: "memory");
    }
    __syncthreads();

    // ---- edge-parallel gather + atomic scatter ----
    const int lane4 = tid & (LANES_PER_EDGE - 1);          // which float4 of the row
    const int egrp  = tid >> 4;                            // edge sub-id within iteration
    constexpr int EDGES_PER_ITER = THREADS / LANES_PER_EDGE;  // 16
    constexpr int NITER = EPB / EDGES_PER_ITER;               // 8

#pragma unroll
    for (int it = 0; it < NITER; ++it) {
        const int local_e = egrp + it * EDGES_PER_ITER;
        const long long ge = e0 + local_e;
        if (ge >= E) break;
        const int src = (int)s_idx[local_e];          // LDS broadcast read
        const int dst = (int)s_idx[EPB + local_e];

        const float4 v =
            *(const float4*)(x + (long long)src * D_FEAT + lane4 * 4);
        float* o = out + (long long)dst * D_FEAT + lane4 * 4;

        // non-returning fp32 atomics -> global_atomic_add_f32
        (void)__hip_atomic_fetch_add(o + 0, v.x, __ATOMIC_RELAXED, __HIP_MEMORY_SCOPE_AGENT);
        (void)__hip_atomic_fetch_add(o + 1, v.y, __ATOMIC_RELAXED, __HIP_MEMORY_SCOPE_AGENT);
        (void)__hip_atomic_fetch_add(o + 2, v.z, __ATOMIC_RELAXED, __HIP_MEMORY_SCOPE_AGENT);
        (void)__hip_atomic_fetch_add(o + 3, v.w, __ATOMIC_RELAXED, __HIP_MEMORY_SCOPE_AGENT);
    }
}

// ---------------------------------------------------------------------------
// Launch
// ---------------------------------------------------------------------------
extern "C" void kernel_launch(void* const* d_in, const int* in_sizes, int n_in,
                              void* d_out, int out_size, void* d_ws, size_t ws_size,
                              hipStream_t stream) {
    const float* x          = (const float*)d_in[0];       // [N, 64] fp32
    const long long* eidx   = (const long long*)d_in[1];   // [2, E] int64
    float* out              = (float*)d_out;               // [N, 64] fp32

    const int E  = in_sizes[1] / 2;
    const int n4 = out_size / 4;

    gnn_zero_kernel<<<(n4 + THREADS - 1) / THREADS, THREADS, 0, stream>>>(
        (float4*)out, n4);

    const int nblocks = (E + EPB - 1) / EPB;
    gnn_scatter_add_kernel<<<nblocks, THREADS, 0, stream>>>(x, eidx, out, E);
}